// SpikingLayer_10591389351936
// MI455X (gfx1250) — compile-verified
//
#include <hip/hip_runtime.h>

typedef __attribute__((ext_vector_type(2))) float v2f;
typedef __attribute__((ext_vector_type(4))) float v4f;
typedef __attribute__((ext_vector_type(8))) float v8f;

#define NN 12288
#define TPB 256
#define WAVES_PER_WG 8
#define KSPAN (NN / WAVES_PER_WG) /* 1536 per wave */
#define ROWS_PER_WG 16
#define THRESH 1.0f

// D = A(16x4 f32) * B(4x16 f32) + C(16x16 f32), B broadcast across columns
// => every column of D holds the 16 partial row dot-products.
__global__ __launch_bounds__(TPB) void spiking_wmma_kernel(
    const float* __restrict__ x,
    const float* __restrict__ W,
    const float* __restrict__ f,
    const float* __restrict__ v,
    float* __restrict__ out)
{
    __shared__ float f_lds[NN];          // 48 KB: whole is_firing vector
    __shared__ float acc[ROWS_PER_WG];   // cross-wave K reduction

    const int tid  = threadIdx.x;
    const int lane = tid & 31;
    const int wave = tid >> 5;
    const int row0 = blockIdx.x * ROWS_PER_WG;

    // Stage is_firing into LDS, 128-bit vectorized (12 iters x 256 threads x 16B)
    {
        const v4f* __restrict__ fg = (const v4f*)f;
        v4f* fl = (v4f*)f_lds;
#pragma unroll
        for (int i = 0; i < NN / 4 / TPB; ++i)
            fl[tid + i * TPB] = fg[tid + i * TPB];
    }
    if (tid < ROWS_PER_WG) acc[tid] = 0.0f;
    __syncthreads();

    // ---- WMMA matvec over this wave's K span ----------------------------
    // A-matrix 16x4 f32 layout: lanes 0-15 = rows M=0..15 holding (K, K+1),
    // lanes 16-31 = rows M=0..15 holding (K+2, K+3).
    const int m     = lane & 15;
    const int khalf = (lane >> 4) * 2;
    const int kbase = wave * KSPAN;
    const float* __restrict__ wrow =
        W + (size_t)(row0 + m) * NN + kbase + khalf;

    v8f c = {0.f, 0.f, 0.f, 0.f, 0.f, 0.f, 0.f, 0.f};
    const bool lo = (lane < 16);

    for (int k = 0; k < KSPAN; k += 16) {
        // 4 x b64 loads of W per lane: 16 row streams x 16B per wave-step
        v2f a0 = *(const v2f*)(wrow + k + 0);
        v2f a1 = *(const v2f*)(wrow + k + 4);
        v2f a2 = *(const v2f*)(wrow + k + 8);
        v2f a3 = *(const v2f*)(wrow + k + 12);

        // B fragments from LDS: f[k..k+3] broadcast across N.
        // B 4x16 layout: VGPR0 = K0 (lanes 0-15) / K2 (lanes 16-31),
        //                VGPR1 = K1 / K3.
        v4f f0 = *(const v4f*)&f_lds[kbase + k + 0];
        v4f f1 = *(const v4f*)&f_lds[kbase + k + 4];
        v4f f2 = *(const v4f*)&f_lds[kbase + k + 8];
        v4f f3 = *(const v4f*)&f_lds[kbase + k + 12];
        v2f b0 = { lo ? f0.x : f0.z, lo ? f0.y : f0.w };
        v2f b1 = { lo ? f1.x : f1.z, lo ? f1.y : f1.w };
        v2f b2 = { lo ? f2.x : f2.z, lo ? f2.y : f2.w };
        v2f b3 = { lo ? f3.x : f3.z, lo ? f3.y : f3.w };

        c = __builtin_amdgcn_wmma_f32_16x16x4_f32(false, a0, false, b0,
                                                  (short)0, c, false, false);
        c = __builtin_amdgcn_wmma_f32_16x16x4_f32(false, a1, false, b1,
                                                  (short)0, c, false, false);
        c = __builtin_amdgcn_wmma_f32_16x16x4_f32(false, a2, false, b2,
                                                  (short)0, c, false, false);
        c = __builtin_amdgcn_wmma_f32_16x16x4_f32(false, a3, false, b3,
                                                  (short)0, c, false, false);
    }

    // ---- Extract the 16 row partials from the C/D fragment --------------
    // D layout (16x16 f32): VGPR r, lanes 0-15 -> D[r, n]; lanes 16-31 -> D[8+r, n].
    // All columns equal, so: lane l in [0,8)  supplies row l   via c[l],
    //                        lane l in [16,24) supplies row l-8 via c[l-16].
    const int sel = lane & 7;
    float val = c[0];
    if (sel == 1) val = c[1];
    else if (sel == 2) val = c[2];
    else if (sel == 3) val = c[3];
    else if (sel == 4) val = c[4];
    else if (sel == 5) val = c[5];
    else if (sel == 6) val = c[6];
    else if (sel == 7) val = c[7];

    const bool writer   = (lane & 15) < 8;
    const int row_local = (lane < 16) ? lane : (lane - 8);
    if (writer) atomicAdd(&acc[row_local], val);   // ds_add_f32
    __syncthreads();

    // ---- LIF update + all four outputs ----------------------------------
    if (tid < ROWS_PER_WG) {
        const int row = row0 + tid;
        const float nx   = acc[tid];
        const float va   = v[row] + nx;
        const bool  fire = va > THRESH;
        out[row]          = x[row];                 // x passthrough
        out[NN + row]     = fire ? 1.0f : 0.0f;     // is_firing_new
        out[2 * NN + row] = fire ? 0.0f : va;       // v_new
        out[3 * NN + row] = nx;                     // node_x
    }
}

extern "C" void kernel_launch(void* const* d_in, const int* in_sizes, int n_in,
                              void* d_out, int out_size, void* d_ws, size_t ws_size,
                              hipStream_t stream) {
    (void)in_sizes; (void)n_in; (void)d_ws; (void)ws_size; (void)out_size;
    const float* x = (const float*)d_in[0];
    const float* W = (const float*)d_in[1];
    const float* f = (const float*)d_in[2];
    const float* v = (const float*)d_in[3];
    float* out = (float*)d_out;

    dim3 grid(NN / ROWS_PER_WG);   // 768 workgroups, 8 waves each
    spiking_wmma_kernel<<<grid, TPB, 0, stream>>>(x, W, f, v, out);
}